// FraudGraphSAGE_2757369004140
// MI455X (gfx1250) — compile-verified
//
#include <hip/hip_runtime.h>
#include <math.h>

// CDNA5 / gfx1250 GraphSAGE (2x SAGEConv + sigmoid head).
// f32 WMMA (V_WMMA_F32_16X16X4_F32) for all four 128x128 GEMMs; edge
// aggregation via global_atomic_add_f32 (entire feature table is L2-resident
// on MI455X's 192MB L2, so scatter traffic never touches HBM).
// Each GEMM wave computes MT=4 M-tiles so every B-operand load is reused 4x.

typedef float v2f __attribute__((ext_vector_type(2)));
typedef float v8f __attribute__((ext_vector_type(8)));

#define C_FEAT 128
#define MT 4            // M-tiles per wave in the GEMM kernels

// ---------------------------------------------------------------- utilities
__global__ void zero_f32(float* __restrict__ p, int n) {
  int i = blockIdx.x * blockDim.x + threadIdx.x;
  if (i < n) p[i] = 0.0f;
}

__global__ void degree_k(const int* __restrict__ dst, float* __restrict__ deg, int E) {
  int e = blockIdx.x * blockDim.x + threadIdx.x;
  if (e < E) atomicAdd(&deg[dst[e]], 1.0f);
}

__global__ void invdeg_k(float* __restrict__ deg, int N) {
  int i = blockIdx.x * blockDim.x + threadIdx.x;
  if (i < N) deg[i] = 1.0f / fmaxf(deg[i], 1.0f);
}

// One wave per edge: lane l moves channels [4l, 4l+4) of the 128-wide row.
__global__ void scatter_add_k(const float* __restrict__ feat,
                              const int* __restrict__ src,
                              const int* __restrict__ dst,
                              float* __restrict__ agg, int E) {
  int e = blockIdx.x * 8 + (threadIdx.x >> 5);
  if (e >= E) return;
  int lane = threadIdx.x & 31;
  int s = src[e], d = dst[e];
  const float4 v = *(const float4*)(feat + (size_t)s * C_FEAT + lane * 4);
  float* p = agg + (size_t)d * C_FEAT + lane * 4;
  atomicAdd(p + 0, v.x);
  atomicAdd(p + 1, v.y);
  atomicAdd(p + 2, v.z);
  atomicAdd(p + 3, v.w);
}

// ------------------------------------------------------------ WMMA tile core
// Computes MT 16x16 tiles of:  scale(Aagg, invdeg) @ Wl + Aself @ Wr
// A layout (16x4 f32): lanes 0-15 -> K=0(v0),K=1(v1); lanes 16-31 -> K=2,K=3.
// B layout (4x16 f32): v0 -> K=0 (lanes 0-15) / K=2 (lanes 16-31); v1 -> K=1/K=3.
// B loads are shared across the MT tiles (4x reuse).
__device__ __forceinline__ void sage_tiles(const float* __restrict__ Aagg,
                                           const float* __restrict__ Aself,
                                           const float* __restrict__ Wl,
                                           const float* __restrict__ Wr,
                                           const float* __restrict__ invdeg,
                                           int m0, int n0, int N, v8f c[MT]) {
  const int lane = threadIdx.x & 31;
  const int l16  = lane & 15;
  const int half = lane >> 4;             // 0: K pair {0,1}, 1: K pair {2,3}
  const int col  = n0 + l16;

  int   row[MT];
  float scl[MT];
#pragma unroll
  for (int t = 0; t < MT; ++t) {
    int r = m0 + t * 16 + l16;
    row[t] = (r < N) ? r : (N - 1);       // clamp (harmless: tail rows unguarded loads only)
    scl[t] = invdeg[row[t]];              // fold mean's 1/deg into A-load
    c[t]   = (v8f){};
  }

#pragma unroll 4
  for (int k0 = 0; k0 < C_FEAT; k0 += 4) {
    const int ka = k0 + half * 2;
    v2f bL, bR;
    bL.x = Wl[(size_t)ka * C_FEAT + col];
    bL.y = Wl[(size_t)(ka + 1) * C_FEAT + col];
    bR.x = Wr[(size_t)ka * C_FEAT + col];
    bR.y = Wr[(size_t)(ka + 1) * C_FEAT + col];
#pragma unroll
    for (int t = 0; t < MT; ++t) {
      const float* ar = Aagg  + (size_t)row[t] * C_FEAT;
      const float* xr = Aself + (size_t)row[t] * C_FEAT;
      v2f aA, aS;
      aA.x = ar[ka]     * scl[t];
      aA.y = ar[ka + 1] * scl[t];
      aS.x = xr[ka];
      aS.y = xr[ka + 1];
      c[t] = __builtin_amdgcn_wmma_f32_16x16x4_f32(false, aA, false, bL, (short)0, c[t], false, false);
      c[t] = __builtin_amdgcn_wmma_f32_16x16x4_f32(false, aS, false, bR, (short)0, c[t], false, false);
    }
  }
}

// Layer 1: h = relu(mean_agg @ W1l + x @ W1r + b1), stored to h1.
__global__ void sage_gemm_relu(const float* __restrict__ agg,
                               const float* __restrict__ xin,
                               const float* __restrict__ Wl,
                               const float* __restrict__ Wr,
                               const float* __restrict__ bias,
                               const float* __restrict__ invdeg,
                               float* __restrict__ hout, int N) {
  const int m0 = blockIdx.x * (16 * MT);
  if (m0 >= N) return;
  const int n0 = (threadIdx.x >> 5) * 16;          // 8 waves -> 8 column tiles
  v8f c[MT];
  sage_tiles(agg, xin, Wl, Wr, invdeg, m0, n0, N, c);

  const int lane = threadIdx.x & 31;
  const int l16  = lane & 15;
  const int half = lane >> 4;
  const float bn = bias[n0 + l16];
#pragma unroll
  for (int t = 0; t < MT; ++t) {
    const int mb = m0 + t * 16;
    if (mb >= N) break;                            // wave-uniform
    float* orow = hout + (size_t)(mb + half * 8) * C_FEAT + n0 + l16;
    if (mb + 16 <= N) {                            // wave-uniform: fast path
#pragma unroll
      for (int v = 0; v < 8; ++v) {
        float val = c[t][v] + bn;
        orow[(size_t)v * C_FEAT] = val > 0.0f ? val : 0.0f;
      }
    } else {                                       // ragged tail
#pragma unroll
      for (int v = 0; v < 8; ++v) {
        const int m = mb + half * 8 + v;
        float val = c[t][v] + bn;
        if (m < N) hout[(size_t)m * C_FEAT + n0 + l16] = val > 0.0f ? val : 0.0f;
      }
    }
  }
}

// Layer 2 fused with FC head: tile = relu(agg@W2l + h1@W2r + b2);
// outacc[m] += sum_n tile[m][n] * Wfc[n]   (half-wave shfl reduction + atomic).
__global__ void sage_gemm_relu_dot(const float* __restrict__ agg,
                                   const float* __restrict__ hin,
                                   const float* __restrict__ Wl,
                                   const float* __restrict__ Wr,
                                   const float* __restrict__ bias,
                                   const float* __restrict__ invdeg,
                                   const float* __restrict__ Wfc,
                                   float* __restrict__ outacc, int N) {
  const int m0 = blockIdx.x * (16 * MT);
  if (m0 >= N) return;
  const int n0 = (threadIdx.x >> 5) * 16;
  v8f c[MT];
  sage_tiles(agg, hin, Wl, Wr, invdeg, m0, n0, N, c);

  const int lane = threadIdx.x & 31;
  const int l16  = lane & 15;
  const int half = lane >> 4;
  const float bn = bias[n0 + l16];
  const float wf = Wfc[n0 + l16];                  // Wfc is [128,1]
#pragma unroll
  for (int t = 0; t < MT; ++t) {
    const int mb = m0 + t * 16;
    if (mb >= N) break;                            // wave-uniform
#pragma unroll
    for (int v = 0; v < 8; ++v) {
      float val = c[t][v] + bn;
      val = val > 0.0f ? val : 0.0f;
      float p = val * wf;
      // reduce across the 16 lanes of this half-wave (xor masks keep bit4 fixed)
      p += __shfl_xor(p, 1, 32);
      p += __shfl_xor(p, 2, 32);
      p += __shfl_xor(p, 4, 32);
      p += __shfl_xor(p, 8, 32);
      const int m = mb + half * 8 + v;
      if (l16 == 0 && m < N) atomicAdd(&outacc[m], p);
    }
  }
}

__global__ void sigmoid_k(const float* __restrict__ acc,
                          const float* __restrict__ bfc,
                          float* __restrict__ out, int N) {
  int i = blockIdx.x * blockDim.x + threadIdx.x;
  if (i < N) out[i] = 1.0f / (1.0f + __expf(-(acc[i] + bfc[0])));
}

// ------------------------------------------------------------------- launch
extern "C" void kernel_launch(void* const* d_in, const int* in_sizes, int n_in,
                              void* d_out, int out_size, void* d_ws, size_t ws_size,
                              hipStream_t stream) {
  const float* x   = (const float*)d_in[0];
  const int*   ei  = (const int*)d_in[1];
  const float* W1l = (const float*)d_in[2];
  const float* W1r = (const float*)d_in[3];
  const float* b1  = (const float*)d_in[4];
  const float* W2l = (const float*)d_in[5];
  const float* W2r = (const float*)d_in[6];
  const float* b2  = (const float*)d_in[7];
  const float* Wfc = (const float*)d_in[8];
  const float* bfc = (const float*)d_in[9];

  const int N = in_sizes[0] / C_FEAT;
  const int E = in_sizes[1] / 2;
  const int* src = ei;
  const int* dst = ei + E;

  float* agg    = (float*)d_ws;                        // [N,128]
  float* h1     = agg + (size_t)N * C_FEAT;            // [N,128]
  float* deg    = h1  + (size_t)N * C_FEAT;            // [N] (becomes invdeg)
  float* outacc = deg + N;                             // [N]

  const int nfeat  = N * C_FEAT;
  const int mtiles = (N + 16 * MT - 1) / (16 * MT);

  // ---- layer 1
  zero_f32<<<(nfeat + 255) / 256, 256, 0, stream>>>(agg, nfeat);
  zero_f32<<<(2 * N + 255) / 256, 256, 0, stream>>>(deg, 2 * N);  // deg+outacc
  degree_k<<<(E + 255) / 256, 256, 0, stream>>>(dst, deg, E);
  invdeg_k<<<(N + 255) / 256, 256, 0, stream>>>(deg, N);
  scatter_add_k<<<(E + 7) / 8, 256, 0, stream>>>(x, src, dst, agg, E);
  sage_gemm_relu<<<mtiles, 256, 0, stream>>>(agg, x, W1l, W1r, b1, deg, h1, N);

  // ---- layer 2 (+ fused FC head accumulation)
  zero_f32<<<(nfeat + 255) / 256, 256, 0, stream>>>(agg, nfeat);
  scatter_add_k<<<(E + 7) / 8, 256, 0, stream>>>(h1, src, dst, agg, E);
  sage_gemm_relu_dot<<<mtiles, 256, 0, stream>>>(agg, h1, W2l, W2r, b2, deg, Wfc, outacc, N);

  // ---- sigmoid head
  sigmoid_k<<<(N + 255) / 256, 256, 0, stream>>>(outacc, bfc, (float*)d_out, N);
}